// GCN_36429912604777
// MI455X (gfx1250) — compile-verified
//
#include <hip/hip_runtime.h>

#define IN_FEAT  256
#define OUT_FEAT 128

typedef float v2f __attribute__((ext_vector_type(2)));
typedef float v8f __attribute__((ext_vector_type(8)));

// ---------------------------------------------------------------------------
// Phase 1: support = x @ W using V_WMMA_F32_16X16X4_F32 (fp32 matrix pipe).
// One block = 16 node rows; 8 waves each own a 16-wide slice of OUT_FEAT.
// x tile staged in LDS (16x256 fp32 = 16 KB) so 8 waves share one global read.
// ---------------------------------------------------------------------------
__global__ __launch_bounds__(256) void gcn_gemm_kernel(
    const float* __restrict__ x, const float* __restrict__ w,
    float* __restrict__ support) {
  __shared__ float xs[16 * IN_FEAT];

  const int tid  = threadIdx.x;
  const int row0 = blockIdx.x * 16;

  // Cooperative load: 16 rows x 256 cols = 1024 float4, 256 threads x 4 each.
  {
    const float4* src = (const float4*)(x + (size_t)row0 * IN_FEAT);
    float4* dst = (float4*)xs;
#pragma unroll
    for (int i = 0; i < 4; ++i)
      dst[tid + 256 * i] = src[tid + 256 * i];
  }
  __syncthreads();

  const int wave = tid >> 5;    // 0..7 : output column tile
  const int lane = tid & 31;
  const int half = lane >> 4;   // 0/1 : which K-pair / which M-half
  const int mn   = lane & 15;   // A/C row index, B/C col index
  const int c0   = wave * 16;

  v8f acc = {};
  const float* arow = xs + mn * IN_FEAT + 2 * half;   // A operand source
  const float* wcol = w + c0 + mn;                    // B operand column base

#pragma unroll 4
  for (int k = 0; k < IN_FEAT; k += 4) {
    v2f a;                       // A 16x4: lane(m,half) -> K = k+2*half+{0,1}
    a.x = arow[k];
    a.y = arow[k + 1];
    v2f b;                       // B 4x16: v0 -> K = k+2*half, v1 -> +1
    b.x = wcol[(size_t)(k + 2 * half) * OUT_FEAT];
    b.y = wcol[(size_t)(k + 2 * half + 1) * OUT_FEAT];
    acc = __builtin_amdgcn_wmma_f32_16x16x4_f32(
        /*neg_a=*/false, a, /*neg_b=*/false, b,
        /*c_mod=*/(short)0, acc, /*reuse_a=*/false, /*reuse_b=*/false);
  }

  // D layout: VGPR r, lane(half,n) -> D[r + 8*half][n]
  float* outp = support + (size_t)row0 * OUT_FEAT + c0 + mn;
#pragma unroll
  for (int r = 0; r < 8; ++r)
    outp[(size_t)(r + 8 * half) * OUT_FEAT] = acc[r];
}

// ---------------------------------------------------------------------------
// Phase 2: out[i, :] = bias[:]  (float4 broadcast writes)
// ---------------------------------------------------------------------------
__global__ __launch_bounds__(256) void gcn_bias_kernel(
    float* __restrict__ out, const float* __restrict__ bias, int total4) {
  int tid = blockIdx.x * 256 + threadIdx.x;
  if (tid >= total4) return;
  const int f4 = tid & (OUT_FEAT / 4 - 1);  // 32 float4 per node row
  ((float4*)out)[tid] = ((const float4*)bias)[f4];
}

// ---------------------------------------------------------------------------
// Phase 3: out[row[e]] += val[e] * support[col[e]]
// One wave per edge; lane owns 4 features -> coalesced 512B gather per edge,
// hardware global_atomic_add_f32 scatter (support & out are L2-resident).
// ---------------------------------------------------------------------------
__global__ __launch_bounds__(256) void gcn_spmm_kernel(
    const int* __restrict__ erow, const int* __restrict__ ecol,
    const float* __restrict__ eval, const float* __restrict__ support,
    float* __restrict__ out, int n_edges) {
  const long long tid = (long long)blockIdx.x * 256 + threadIdx.x;
  const int e = (int)(tid >> 5);
  if (e >= n_edges) return;
  const int f = ((int)tid & 31) * 4;

  const int   col = ecol[e];
  const int   row = erow[e];
  const float v   = eval[e];

  const float4 s = *(const float4*)(support + (size_t)col * OUT_FEAT + f);
  float* dst = out + (size_t)row * OUT_FEAT + f;
  unsafeAtomicAdd(dst + 0, s.x * v);
  unsafeAtomicAdd(dst + 1, s.y * v);
  unsafeAtomicAdd(dst + 2, s.z * v);
  unsafeAtomicAdd(dst + 3, s.w * v);
}

// ---------------------------------------------------------------------------
extern "C" void kernel_launch(void* const* d_in, const int* in_sizes, int n_in,
                              void* d_out, int out_size, void* d_ws, size_t ws_size,
                              hipStream_t stream) {
  const float* x      = (const float*)d_in[0];  // [N, 256]
  const int*   erow   = (const int*)  d_in[1];  // [E]
  const int*   ecol   = (const int*)  d_in[2];  // [E]
  const float* eval   = (const float*)d_in[3];  // [E]
  const float* weight = (const float*)d_in[4];  // [256, 128]
  const float* bias   = (const float*)d_in[5];  // [128]
  float*       out    = (float*)d_out;          // [N, 128]

  const int n_nodes = in_sizes[0] / IN_FEAT;    // 100000
  const int n_edges = in_sizes[1];              // 1600000

  float* support = (float*)d_ws;                // [N, 128] scratch (51.2 MB)

  // Phase 1: dense GEMM via fp32 WMMA. N is a multiple of 16 (100000/16=6250).
  gcn_gemm_kernel<<<n_nodes / 16, 256, 0, stream>>>(x, weight, support);

  // Phase 2: bias broadcast into out.
  const int total4 = n_nodes * (OUT_FEAT / 4);
  gcn_bias_kernel<<<(total4 + 255) / 256, 256, 0, stream>>>(out, bias, total4);

  // Phase 3: edge gather-scale-scatter with hw f32 atomics.
  const long long spmm_threads = (long long)n_edges * 32;
  const int spmm_blocks = (int)((spmm_threads + 255) / 256);
  gcn_spmm_kernel<<<spmm_blocks, 256, 0, stream>>>(erow, ecol, eval, support,
                                                   out, n_edges);
}